// SparseAttention_41137196761698
// MI455X (gfx1250) — compile-verified
//
#include <hip/hip_runtime.h>
#include <hip/hip_bf16.h>

// ---------------------------------------------------------------------------
// Sparse neighborhood attention for MI455X (gfx1250, wave32).
//   B=2, P=4096, D=128, H=4, HD=32, neighbors per row <= 17 (cap 32).
// Pipeline: mask->index lists, QKV f32-WMMA GEMMs, wave-per-row sparse
// attention, output f32-WMMA GEMM.
//
// GEMM v2: 64-row M-tile per block, 4 accumulators per wave so each W
// (B-matrix) fragment load feeds 4 v_wmma_f32_16x16x4_f32 ops. This cuts
// per-GEMM W re-read traffic from 32 MB to 8 MB of L2 and raises the
// wmma : global-load ratio from 0.5 to 4.
// ---------------------------------------------------------------------------

typedef __attribute__((ext_vector_type(2))) float v2f;
typedef __attribute__((ext_vector_type(8))) float v8f;

#define PB 4096      // P
#define DB 128       // D
#define NH 4         // heads
#define HD 32        // head dim
#define NB_MAX 32    // neighbor cap (true max is 17)
#define MT 64        // M rows per GEMM block (4 x 16-row WMMA tiles)
#define LDS_STRIDE 132  // 128 + 4 pad: breaks 64-bank conflict on column reads

// ---------------------------------------------------------------------------
// Kernel 1: compact boolean mask rows into index lists (wave per row).
// Mask scan is the single largest memory stream (16 MB) -> 4B/lane loads.
// ---------------------------------------------------------------------------
__global__ __launch_bounds__(256) void build_nbrs(
    const unsigned char* __restrict__ mask, int* __restrict__ nbidx,
    int* __restrict__ nbcnt) {
  const int wid  = threadIdx.x >> 5;
  const int lane = threadIdx.x & 31;
  const int p = blockIdx.x * 8 + wid;
  if (p >= PB) return;

  int base = 0;
  const unsigned prefix_mask = (lane == 0) ? 0u : (~0u >> (32 - lane));
  for (int c0 = 0; c0 < PB; c0 += 128) {
    unsigned word =
        *(const unsigned*)(mask + (size_t)p * PB + c0 + lane * 4);
    #pragma unroll
    for (int t = 0; t < 4; ++t) {
      bool m = ((word >> (8 * t)) & 0xffu) != 0u;
      unsigned bal = (unsigned)__ballot(m);   // wave32: low 32 bits
      if (m) {
        int slot = base + __popc(bal & prefix_mask);
        if (slot < NB_MAX) nbidx[p * NB_MAX + slot] = c0 + lane * 4 + t;
      }
      base += __popc(bal);
    }
  }
  if (lane == 0) nbcnt[p] = base < NB_MAX ? base : NB_MAX;
}

// ---------------------------------------------------------------------------
// Kernel 2: Y[m, e] = sum_d X[m, d] * W[e, d]    (M x 128 x 128, f32 WMMA)
// Block: 256 threads = 8 waves, owns a 64x128 X tile in LDS. Wave w owns
// N-tile [16w, 16w+16) across all 4 M-subtiles: per k-step one B fragment
// (global) is reused by 4 A fragments (LDS) -> 4 WMMAs.
// Fragment layouts per CDNA5 ISA 7.12.2 (B assumed mirror of A).
// ---------------------------------------------------------------------------
__global__ __launch_bounds__(256) void gemm_xwt(
    const float* __restrict__ X, const float* __restrict__ W,
    float* __restrict__ Y, int M) {
  __shared__ float xs[MT][LDS_STRIDE];

  const int tid  = threadIdx.x;
  const int wid  = tid >> 5;
  const int lane = tid & 31;
  const int row0 = blockIdx.x * MT;

  // Cooperative load of the 64x128 X tile (32 elements per thread).
  #pragma unroll
  for (int i = tid; i < MT * DB; i += 256) {
    const int m = i >> 7, kk = i & 127;
    xs[m][kk] = (row0 + m < M) ? X[(size_t)(row0 + m) * DB + kk] : 0.0f;
  }
  __syncthreads();

  const int mrow  = lane & 15;       // A: M index, B/C/D: N index
  const int khalf = lane >> 4;       // which K pair this half-wave holds
  const int n0    = wid * 16;

  v8f c0 = {}, c1 = {}, c2 = {}, c3 = {};
  #pragma unroll
  for (int kb = 0; kb < DB; kb += 4) {
    const int ka = kb + 2 * khalf;                          // even -> 8B ok
    const v2f b = *(const v2f*)&W[(size_t)(n0 + mrow) * DB + ka];
    const v2f a0 = *(const v2f*)&xs[ 0 + mrow][ka];
    const v2f a1 = *(const v2f*)&xs[16 + mrow][ka];
    const v2f a2 = *(const v2f*)&xs[32 + mrow][ka];
    const v2f a3 = *(const v2f*)&xs[48 + mrow][ka];
    c0 = __builtin_amdgcn_wmma_f32_16x16x4_f32(false, a0, false, b, (short)0,
                                               c0, false, false);
    c1 = __builtin_amdgcn_wmma_f32_16x16x4_f32(false, a1, false, b, (short)0,
                                               c1, false, false);
    c2 = __builtin_amdgcn_wmma_f32_16x16x4_f32(false, a2, false, b, (short)0,
                                               c2, false, false);
    c3 = __builtin_amdgcn_wmma_f32_16x16x4_f32(false, a3, false, b, (short)0,
                                               c3, false, false);
  }

  // C/D layout: vgpr g -> M = g + 8*khalf (within 16-row subtile), N = mrow.
  #pragma unroll
  for (int g = 0; g < 8; ++g) {
    const int mbase = row0 + g + 8 * khalf;
    const int col = n0 + mrow;
    if (mbase      < M) Y[(size_t)(mbase     ) * DB + col] = c0[g];
    if (mbase + 16 < M) Y[(size_t)(mbase + 16) * DB + col] = c1[g];
    if (mbase + 32 < M) Y[(size_t)(mbase + 32) * DB + col] = c2[g];
    if (mbase + 48 < M) Y[(size_t)(mbase + 48) * DB + col] = c3[g];
  }
}

// ---------------------------------------------------------------------------
// Kernel 3: sparse attention. One wave per (b, p); lane == head-dim element
// (HD == 32 == wave32). Scores via coalesced 32-lane dots + shfl_xor tree;
// register-resident softmax (score parked in lane j), weighted V gather.
// ---------------------------------------------------------------------------
__global__ __launch_bounds__(256) void sparse_attn(
    const float* __restrict__ q, const float* __restrict__ k,
    const float* __restrict__ v, const int* __restrict__ nbidx,
    const int* __restrict__ nbcnt, float* __restrict__ ctx) {
  const int wid  = threadIdx.x >> 5;
  const int lane = threadIdx.x & 31;
  const int row  = blockIdx.x * 8 + wid;   // [0, B*P)
  const int b    = row >> 12;              // P = 4096
  const int p    = row & (PB - 1);

  const int cnt = nbcnt[p];
  const float scale = 0.17677669529663687f;  // 1/sqrt(HD)
  const float NEG_INF = -__builtin_inff();

  #pragma unroll
  for (int h = 0; h < NH; ++h) {
    const size_t qoff = (size_t)row * DB + h * HD;
    const float ql = q[qoff + lane];

    // Pass 1: neighbor scores; lane j ends up holding score_j.
    float s = NEG_INF;
    for (int j = 0; j < cnt; ++j) {
      const int nb = nbidx[p * NB_MAX + j];
      const size_t koff = ((size_t)(b * PB + nb)) * DB + h * HD;
      float t = ql * k[koff + lane];
      #pragma unroll
      for (int off = 16; off > 0; off >>= 1) t += __shfl_xor(t, off, 32);
      if (lane == j) s = t * scale;
    }

    // Softmax across lanes (empty row -> all-zero weights, matches
    // nan_to_num in the reference).
    float mx = s;
    #pragma unroll
    for (int off = 16; off > 0; off >>= 1)
      mx = fmaxf(mx, __shfl_xor(mx, off, 32));
    float e = (lane < cnt) ? __expf(s - mx) : 0.0f;
    float sum = e;
    #pragma unroll
    for (int off = 16; off > 0; off >>= 1) sum += __shfl_xor(sum, off, 32);
    const float w = (sum > 0.0f) ? (e / sum) : 0.0f;

    // Pass 2: ctx[lane] = sum_j w_j * V[nb_j][h][lane].
    float acc = 0.0f;
    for (int j = 0; j < cnt; ++j) {
      const int nb = nbidx[p * NB_MAX + j];
      const float wj = __shfl(w, j, 32);
      acc += wj * v[((size_t)(b * PB + nb)) * DB + h * HD + lane];
    }
    ctx[(size_t)row * DB + h * HD + lane] = acc;
  }
}

// ---------------------------------------------------------------------------
// Host-side launcher.
// Inputs: x[B,P,D] f32, Wq/Wk/Wv/Wo[D,D] f32, mask[P,P] bool(1B).
// Workspace: q,k,v,ctx (4 x 4 MB f32) + nbidx (512 KB) + nbcnt (16 KB).
// ---------------------------------------------------------------------------
extern "C" void kernel_launch(void* const* d_in, const int* in_sizes, int n_in,
                              void* d_out, int out_size, void* d_ws,
                              size_t ws_size, hipStream_t stream) {
  (void)in_sizes; (void)n_in; (void)out_size; (void)ws_size;

  const float* x  = (const float*)d_in[0];
  const float* Wq = (const float*)d_in[1];
  const float* Wk = (const float*)d_in[2];
  const float* Wv = (const float*)d_in[3];
  const float* Wo = (const float*)d_in[4];
  const unsigned char* mask = (const unsigned char*)d_in[5];
  float* out = (float*)d_out;

  const int M = 2 * PB;                 // B*P = 8192 rows, batches contiguous
  float* q   = (float*)d_ws;            // M*D
  float* k   = q   + (size_t)M * DB;
  float* v   = k   + (size_t)M * DB;
  float* ctx = v   + (size_t)M * DB;
  int* nbidx = (int*)(ctx + (size_t)M * DB);
  int* nbcnt = nbidx + PB * NB_MAX;

  build_nbrs<<<PB / 8, 256, 0, stream>>>(mask, nbidx, nbcnt);

  gemm_xwt<<<M / MT, 256, 0, stream>>>(x, Wq, q, M);
  gemm_xwt<<<M / MT, 256, 0, stream>>>(x, Wk, k, M);
  gemm_xwt<<<M / MT, 256, 0, stream>>>(x, Wv, v, M);

  sparse_attn<<<M / 8, 256, 0, stream>>>(q, k, v, nbidx, nbcnt, ctx);

  gemm_xwt<<<M / MT, 256, 0, stream>>>(ctx, Wo, out, M);
}